// prediction_net_1125281432313
// MI455X (gfx1250) — compile-verified
//
#include <hip/hip_runtime.h>
#include <math.h>

typedef __attribute__((ext_vector_type(2))) float v2f;
typedef __attribute__((ext_vector_type(8))) float v8f;

#define LN_EPS 1e-5f
#define SQ_EPS 1e-9f

__device__ __forceinline__ float gelu_exact(float x) {
    return 0.5f * x * (1.0f + erff(x * 0.70710678118654752440f));
}

__device__ __forceinline__ v8f wmma4(v2f a, v2f b, v8f c) {
    // D(16x16) += A(16x4) * B(4x16), fp32 matrix core
    return __builtin_amdgcn_wmma_f32_16x16x4_f32(false, a, false, b, (short)0, c, false, false);
}

__device__ __forceinline__ v8f vzero8() {
    v8f z = {0.f,0.f,0.f,0.f,0.f,0.f,0.f,0.f};
    return z;
}

__global__ __attribute__((amdgpu_waves_per_eu(1))) __launch_bounds__(256)
void capsnet_fused(
    const float* __restrict__ x,   const float* __restrict__ W1, const float* __restrict__ b1,
    const float* __restrict__ W2,  const float* __restrict__ b2,
    const float* __restrict__ ln_g,const float* __restrict__ ln_b,
    const float* __restrict__ Wcap,const float* __restrict__ Wf, const float* __restrict__ bf,
    float* __restrict__ out)
{
    __shared__ float xh[16*256];    // x tile, later reused for h2   (16 KB)
    __shared__ float h1s[16*32];    // (2 KB)
    __shared__ float mu_s[16];
    __shared__ float rstd_s[16];
    __shared__ float u_t[32*130];   // u transposed: [i][k][b], padded stride 130 (16.6 KB)
    __shared__ float bij[16*513];   // routing logits: b*513 + i*16 + j (32.8 KB)
    __shared__ float cij[16*513];   // routing coeffs, same layout      (32.8 KB)
    __shared__ float part[16*17];   // output partials [j][b]

    const int t    = threadIdx.x;   // 256 threads = 8 waves
    const int lane = t & 31;
    const int wav  = t >> 5;        // wave id: owns output capsules j0=wav, j1=wav+8
    const int bb   = lane & 15;     // b (or d) index within lane half
    const int hi   = lane >> 4;     // lane half

    const int tile = blockIdx.x;    // 16 batch rows per block
    const float* xg = x + (size_t)tile * 16 * 256;

    // ---------------- stage x tile ----------------
    #pragma unroll
    for (int r = 0; r < 16; ++r) xh[t + 256*r] = xg[t + 256*r];
    __syncthreads();

    // ---------------- h1 = gelu(x @ W1 + b1)  [16x32], thread=(b0/b0+8, n) ----------------
    {
        int n = t & 31, b0 = t >> 5;
        float acc0 = b1[n], acc1 = acc0;
        #pragma unroll 8
        for (int k = 0; k < 256; ++k) {
            float w = W1[k*32 + n];
            acc0 = fmaf(xh[b0*256 + k],       w, acc0);
            acc1 = fmaf(xh[(b0+8)*256 + k],   w, acc1);
        }
        h1s[b0*32 + n]     = gelu_exact(acc0);
        h1s[(b0+8)*32 + n] = gelu_exact(acc1);
    }
    __syncthreads();

    // ---------------- h2 = gelu(h1 @ W2 + b2)  [16x256], overwrites xh ----------------
    {
        int n0 = t & 31, b0 = t >> 5;
        float h1a[32], h1b[32];
        #pragma unroll
        for (int k = 0; k < 32; ++k) { h1a[k] = h1s[b0*32 + k]; h1b[k] = h1s[(b0+8)*32 + k]; }
        #pragma unroll
        for (int rr = 0; rr < 8; ++rr) {
            int n = n0 + 32*rr;
            float acc0 = b2[n], acc1 = acc0;
            #pragma unroll
            for (int k = 0; k < 32; ++k) {
                float w = W2[k*256 + n];
                acc0 = fmaf(h1a[k], w, acc0);
                acc1 = fmaf(h1b[k], w, acc1);
            }
            xh[b0*256 + n]     = gelu_exact(acc0);
            xh[(b0+8)*256 + n] = gelu_exact(acc1);
        }
    }
    __syncthreads();

    // ---------------- LayerNorm stats: wave w handles samples b=w and b=w+8 ----------------
    {
        int ba = wav, bc = wav + 8;
        float sa = 0.f, sb = 0.f;
        #pragma unroll
        for (int r = 0; r < 8; ++r) { sa += xh[ba*256 + lane + 32*r]; sb += xh[bc*256 + lane + 32*r]; }
        #pragma unroll
        for (int m = 16; m >= 1; m >>= 1) { sa += __shfl_xor(sa, m, 32); sb += __shfl_xor(sb, m, 32); }
        float mua = sa * (1.0f/256.0f), mub = sb * (1.0f/256.0f);
        float va = 0.f, vb = 0.f;
        #pragma unroll
        for (int r = 0; r < 8; ++r) {
            float da = xh[ba*256 + lane + 32*r] - mua; va = fmaf(da, da, va);
            float db = xh[bc*256 + lane + 32*r] - mub; vb = fmaf(db, db, vb);
        }
        #pragma unroll
        for (int m = 16; m >= 1; m >>= 1) { va += __shfl_xor(va, m, 32); vb += __shfl_xor(vb, m, 32); }
        if (lane == 0) {
            mu_s[ba] = mua; rstd_s[ba] = rsqrtf(va * (1.0f/256.0f) + LN_EPS);
            mu_s[bc] = mub; rstd_s[bc] = rsqrtf(vb * (1.0f/256.0f) + LN_EPS);
        }
    }
    __syncthreads();

    // ---------------- LN apply + capsule squash -> u_t[i][k][b], thread=(b0/b0+8, i) ----------------
    {
        int i = t & 31, b0 = t >> 5;
        float g8[8], be8[8];
        #pragma unroll
        for (int k = 0; k < 8; ++k) { g8[k] = ln_g[i*8 + k]; be8[k] = ln_b[i*8 + k]; }
        #pragma unroll
        for (int bsel = 0; bsel < 2; ++bsel) {
            int b = b0 + 8*bsel;
            float mu = mu_s[b], rs = rstd_s[b];
            float vals[8]; float nsq = 0.f;
            #pragma unroll
            for (int k = 0; k < 8; ++k) {
                float v = (xh[b*256 + i*8 + k] - mu) * rs * g8[k] + be8[k];
                vals[k] = v; nsq = fmaf(v, v, nsq);
            }
            float scale = nsq / ((1.0f + nsq) * sqrtf(nsq + SQ_EPS));
            #pragma unroll
            for (int k = 0; k < 8; ++k) u_t[i*130 + k*16 + b] = scale * vals[k];
        }
    }

    // ---------------- init routing state: b_ij = 0, c_ij = 1/16 ----------------
    #pragma unroll
    for (int r = 0; r < 32; ++r) { bij[t + 256*r] = 0.f; cij[t + 256*r] = 0.0625f; }
    if (t < 16) { bij[8192 + t] = 0.f; cij[8192 + t] = 0.0625f; }
    __syncthreads();

    // ---------------- A operand cache: Wcap[i][j]^T tiles for j0=wav, j1=wav+8 (256 VGPRs) ----
    // A-layout (f32 16x4): lane: M=d=bb ; VGPR r: K = 2*hi + r (call0), +4 (call1)
    v2f a0A[32], a1A[32], a0B[32], a1B[32];
    #pragma unroll
    for (int i = 0; i < 32; ++i) {
        const float* wcA = Wcap + (size_t)((i*16 + wav    )*8)*16 + bb;  // [i][j][k][d]
        const float* wcB = Wcap + (size_t)((i*16 + wav + 8)*8)*16 + bb;
        a0A[i][0] = wcA[(2*hi+0)*16];  a0A[i][1] = wcA[(2*hi+1)*16];
        a1A[i][0] = wcA[(2*hi+4)*16];  a1A[i][1] = wcA[(2*hi+5)*16];
        a0B[i][0] = wcB[(2*hi+0)*16];  a0B[i][1] = wcB[(2*hi+1)*16];
        a1B[i][0] = wcB[(2*hi+4)*16];  a1B[i][1] = wcB[(2*hi+5)*16];
    }

    // ---------------- dynamic routing: 3 iterations ----------------
    v8f vA = vzero8(), vB = vzero8();
    for (int r = 0; r < 3; ++r) {
        // s_j[d,b] = sum_i c[b,i,j]*u_hat[b,i,j,d] : scale B columns, 4 independent WMMA chains
        v8f sA0 = vzero8(), sA1 = vzero8(), sB0 = vzero8(), sB1 = vzero8();
        #pragma unroll
        for (int i = 0; i < 32; ++i) {
            const float* ub = u_t + i*130 + bb;
            v2f bu0, bu1;
            bu0[0] = ub[(2*hi+0)*16];  bu0[1] = ub[(2*hi+1)*16];
            bu1[0] = ub[(2*hi+4)*16];  bu1[1] = ub[(2*hi+5)*16];
            float cA = cij[bb*513 + i*16 + wav];
            float cB = cij[bb*513 + i*16 + wav + 8];
            v2f p0, p1;
            p0[0] = bu0[0]*cA; p0[1] = bu0[1]*cA; p1[0] = bu1[0]*cA; p1[1] = bu1[1]*cA;
            if (i & 1) { sA1 = wmma4(a0A[i], p0, sA1); sA1 = wmma4(a1A[i], p1, sA1); }
            else       { sA0 = wmma4(a0A[i], p0, sA0); sA0 = wmma4(a1A[i], p1, sA0); }
            p0[0] = bu0[0]*cB; p0[1] = bu0[1]*cB; p1[0] = bu1[0]*cB; p1[1] = bu1[1]*cB;
            if (i & 1) { sB1 = wmma4(a0B[i], p0, sB1); sB1 = wmma4(a1B[i], p1, sB1); }
            else       { sB0 = wmma4(a0B[i], p0, sB0); sB0 = wmma4(a1B[i], p1, sB0); }
        }
        v8f sA = sA0 + sA1, sB = sB0 + sB1;
        // squash over d (d = vgpr row + 8*hi): per-lane 8 + cross-half exchange
        float nA = 0.f, nB = 0.f;
        #pragma unroll
        for (int v = 0; v < 8; ++v) { nA = fmaf(sA[v], sA[v], nA); nB = fmaf(sB[v], sB[v], nB); }
        nA += __shfl_xor(nA, 16, 32);  nB += __shfl_xor(nB, 16, 32);
        float scA = nA / ((1.0f + nA) * sqrtf(nA + SQ_EPS));
        float scB = nB / ((1.0f + nB) * sqrtf(nB + SQ_EPS));
        #pragma unroll
        for (int v = 0; v < 8; ++v) { vA[v] = sA[v]*scA; vB[v] = sB[v]*scB; }

        if (r < 2) {
            // agreement[b,i,j] = <u_hat[b,i,j,:], v_j[b,:]> ; recompute u_hat on matrix core
            #pragma unroll
            for (int i = 0; i < 32; ++i) {
                const float* ub = u_t + i*130 + bb;
                v2f bu0, bu1;
                bu0[0] = ub[(2*hi+0)*16];  bu0[1] = ub[(2*hi+1)*16];
                bu1[0] = ub[(2*hi+4)*16];  bu1[1] = ub[(2*hi+5)*16];
                v8f uA = vzero8(); uA = wmma4(a0A[i], bu0, uA); uA = wmma4(a1A[i], bu1, uA);
                v8f uB = vzero8(); uB = wmma4(a0B[i], bu0, uB); uB = wmma4(a1B[i], bu1, uB);
                float agA = 0.f, agB = 0.f;
                #pragma unroll
                for (int v = 0; v < 8; ++v) { agA = fmaf(uA[v], vA[v], agA); agB = fmaf(uB[v], vB[v], agB); }
                agA += __shfl_xor(agA, 16, 32);  agB += __shfl_xor(agB, 16, 32);
                int ia = bb*513 + i*16 + wav;
                if (lane < 16) { bij[ia] += agA; bij[ia + 8] += agB; }   // exclusive slots
            }
            __syncthreads();
            // softmax over j per (b,i): thread covers (b0,i) and (b0+8,i)
            #pragma unroll
            for (int bsel = 0; bsel < 2; ++bsel) {
                int base = (wav + 8*bsel)*513 + (t & 31)*16;
                float mx = bij[base];
                #pragma unroll
                for (int j = 1; j < 16; ++j) mx = fmaxf(mx, bij[base + j]);
                float e[16]; float sum = 0.f;
                #pragma unroll
                for (int j = 0; j < 16; ++j) { e[j] = expf(bij[base + j] - mx); sum += e[j]; }
                float inv = 1.0f / sum;
                #pragma unroll
                for (int j = 0; j < 16; ++j) cij[base + j] = e[j] * inv;
            }
            __syncthreads();
        }
    }

    // ---------------- out[b] = v.reshape(256) @ Wf + bf ----------------
    {
        float pA = 0.f, pB = 0.f;
        #pragma unroll
        for (int v = 0; v < 8; ++v) {
            pA = fmaf(vA[v], Wf[(wav    )*16 + v + 8*hi], pA);
            pB = fmaf(vB[v], Wf[(wav + 8)*16 + v + 8*hi], pB);
        }
        pA += __shfl_xor(pA, 16, 32);  pB += __shfl_xor(pB, 16, 32);
        if (lane < 16) { part[wav*17 + bb] = pA; part[(wav+8)*17 + bb] = pB; }
    }
    __syncthreads();
    if (t < 16) {
        float acc = bf[0];
        #pragma unroll
        for (int j = 0; j < 16; ++j) acc += part[j*17 + t];   // fixed order -> deterministic
        out[tile*16 + t] = acc;
    }
}

extern "C" void kernel_launch(void* const* d_in, const int* in_sizes, int n_in,
                              void* d_out, int out_size, void* d_ws, size_t ws_size,
                              hipStream_t stream) {
    const float* x    = (const float*)d_in[0];
    const float* W1   = (const float*)d_in[1];
    const float* b1   = (const float*)d_in[2];
    const float* W2   = (const float*)d_in[3];
    const float* b2   = (const float*)d_in[4];
    const float* ln_g = (const float*)d_in[5];
    const float* ln_b = (const float*)d_in[6];
    const float* Wcap = (const float*)d_in[7];
    const float* Wf   = (const float*)d_in[8];
    const float* bf   = (const float*)d_in[9];
    float* out = (float*)d_out;

    const int B = in_sizes[0] / 256;     // 16384
    const int blocks = B / 16;           // 1024 workgroups, 16 samples each
    capsnet_fused<<<blocks, 256, 0, stream>>>(x, W1, b1, W2, b2, ln_g, ln_b, Wcap, Wf, bf, out);
}